// VectorQuantizerEMA_78451872629293
// MI455X (gfx1250) — compile-verified
//
#include <hip/hip_runtime.h>

typedef float v2f __attribute__((ext_vector_type(2)));
typedef float v4f __attribute__((ext_vector_type(4)));
typedef float v8f __attribute__((ext_vector_type(8)));

#define NUM_EMB   1024
#define EMB_DIM   64
#define N_ROWS    (128 * 1024)   // 131072 flattened vectors
#define DECAY     0.99f
#define OMD       0.01f          // 1 - DECAY
#define EPS       1e-5f

// ---------------------------------------------------------------------------
// Zero the accumulator region of the workspace (counts + sums).
// ---------------------------------------------------------------------------
__global__ void vq_zero_ws(float* __restrict__ ws, int n) {
    int i = blockIdx.x * blockDim.x + threadIdx.x;
    if (i < n) ws[i] = 0.0f;
}

// ---------------------------------------------------------------------------
// cnorm[k] = 0.5 * ||codebook[k]||^2   (one wave32 per codebook row)
// ---------------------------------------------------------------------------
__global__ void vq_cnorm(const float* __restrict__ cb, float* __restrict__ cnorm) {
    const int wave = (blockIdx.x * blockDim.x + threadIdx.x) >> 5;
    const int lane = threadIdx.x & 31;
    if (wave >= NUM_EMB) return;
    const float* row = cb + wave * EMB_DIM;
    float a = row[lane];
    float b = row[lane + 32];
    float s = a * a + b * b;
    #pragma unroll
    for (int m = 16; m >= 1; m >>= 1) s += __shfl_xor(s, m, 32);
    if (lane == 0) cnorm[wave] = 0.5f * s;
}

// ---------------------------------------------------------------------------
// Main kernel: one wave32 handles 16 rows of x.
//   scores(16x16 tile) = A(16x64) * B(64x16) via 16 chained
//   v_wmma_f32_16x16x4_f32; argmax(score - 0.5||c||^2) == argmin(dist).
// Then writes one-hot rows (nontemporal), quantized rows, and scatter-adds
// cluster sums / counts.
// ---------------------------------------------------------------------------
__global__ __launch_bounds__(256) void vq_assign(
    const float* __restrict__ x,       // [N_ROWS, 64]
    const float* __restrict__ cb,      // [1024, 64]
    const float* __restrict__ cnorm,   // [1024] = 0.5*||c||^2
    float* __restrict__ disc,          // [N_ROWS, 1024]
    float* __restrict__ quant,         // [N_ROWS, 64]
    float* __restrict__ counts,        // [1024]
    float* __restrict__ sums)          // [1024, 64]
{
    const int lane = threadIdx.x & 31;
    const int wave = (blockIdx.x * blockDim.x + threadIdx.x) >> 5;
    const int m0   = wave * 16;
    const int l16  = lane & 15;
    const int hi   = lane >> 4;        // 0: lanes 0-15, 1: lanes 16-31
    const int koff = hi * 2;           // ISA A/B layout: K pair 0/1 vs 2/3

    // ---- Preload A tile (16 rows x K=64) in WMMA A-layout: lane holds row
    // m0+l16, float2 per K-quad.
    v2f a[16];
    {
        const float* xr = x + (size_t)(m0 + l16) * EMB_DIM + koff;
        #pragma unroll
        for (int j = 0; j < 16; ++j) {
            a[j].x = xr[4 * j];
            a[j].y = xr[4 * j + 1];
        }
    }

    float bestv[8];
    int   bestn[8];
    #pragma unroll
    for (int v = 0; v < 8; ++v) { bestv[v] = -3.4e38f; bestn[v] = 0; }

    // ---- Sweep all 1024 codes in 16-wide tiles.
    for (int n0 = 0; n0 < NUM_EMB; n0 += 16) {
        const float* crow = cb + (size_t)(n0 + l16) * EMB_DIM + koff;
        const float  cn   = cnorm[n0 + l16];   // column-dependent bias

        v8f acc = {};
        #pragma unroll
        for (int j = 0; j < 16; ++j) {
            v2f b;
            b.x = crow[4 * j];          // B[k][n] = codebook[n0+n][k]
            b.y = crow[4 * j + 1];
            acc = __builtin_amdgcn_wmma_f32_16x16x4_f32(
                false, a[j], false, b, (short)0, acc, false, false);
        }

        // acc[v] @ lane: row M = v + 8*hi, col N = n0 + l16
        #pragma unroll
        for (int v = 0; v < 8; ++v) {
            float s = acc[v] - cn;
            if (s > bestv[v]) { bestv[v] = s; bestn[v] = n0 + l16; }
        }
    }

    // ---- Reduce argmax across the 16 column-lanes of each half-wave.
    // Lower index wins ties (matches argmin-first semantics).
    #pragma unroll
    for (int v = 0; v < 8; ++v) {
        #pragma unroll
        for (int mask = 1; mask <= 8; mask <<= 1) {
            float ov = __shfl_xor(bestv[v], mask, 32);
            int   on = __shfl_xor(bestn[v], mask, 32);
            if (ov > bestv[v] || (ov == bestv[v] && on < bestn[v])) {
                bestv[v] = ov; bestn[v] = on;
            }
        }
    }

    // ---- Broadcast the winning index of each of the 16 rows to all lanes.
    int rowIdx[16];
    #pragma unroll
    for (int r = 0; r < 16; ++r)
        rowIdx[r] = __shfl(bestn[r & 7], (r < 8) ? 0 : 16, 32);

    // ---- Outputs.
    #pragma unroll 1
    for (int r = 0; r < 16; ++r) {
        const int row = m0 + r;
        const int bn  = rowIdx[r];

        // one-hot row: 1024 floats, nontemporal 16B stores (streaming 537MB)
        float* drow = disc + (size_t)row * NUM_EMB;
        #pragma unroll
        for (int c = 0; c < 8; ++c) {
            const int base = c * 128 + lane * 4;
            v4f val;
            val.x = (bn == base    ) ? 1.0f : 0.0f;
            val.y = (bn == base + 1) ? 1.0f : 0.0f;
            val.z = (bn == base + 2) ? 1.0f : 0.0f;
            val.w = (bn == base + 3) ? 1.0f : 0.0f;
            __builtin_nontemporal_store(val, (v4f*)(drow + base));
        }

        // quantized row = codebook[bn] (codebook stays hot in L2)
        const int d = lane * 2;
        v2f q = *(const v2f*)(cb + (size_t)bn * EMB_DIM + d);
        __builtin_nontemporal_store(q, (v2f*)(quant + (size_t)row * EMB_DIM + d));

        // scatter-add x row into cluster sums; count once per row
        const float* xr = x + (size_t)row * EMB_DIM;
        atomicAdd(&sums[bn * EMB_DIM + d],     xr[d]);
        atomicAdd(&sums[bn * EMB_DIM + d + 1], xr[d + 1]);
        if (lane == 0) atomicAdd(&counts[bn], 1.0f);
    }
}

// ---------------------------------------------------------------------------
// new_count = EMA + Laplace normalization (single block of 1024 threads)
// ---------------------------------------------------------------------------
__global__ void vq_count(const float* __restrict__ ema_count,
                         const float* __restrict__ counts,
                         float* __restrict__ out_count) {
    __shared__ float red[NUM_EMB];
    const int k = threadIdx.x;
    const float nc = ema_count[k] * DECAY + counts[k] * OMD;
    red[k] = nc;
    __syncthreads();
    for (int s = NUM_EMB / 2; s >= 1; s >>= 1) {
        if (k < s) red[k] += red[k + s];
        __syncthreads();
    }
    const float n = red[0];
    out_count[k] = (nc + EPS) / (n + NUM_EMB * EPS) * n;
}

// ---------------------------------------------------------------------------
// new_weight = EMA; new_codebook = new_weight / new_count
// ---------------------------------------------------------------------------
__global__ void vq_weight(const float* __restrict__ ema_weight,
                          const float* __restrict__ sums,
                          const float* __restrict__ new_count,
                          float* __restrict__ out_weight,
                          float* __restrict__ out_codebook) {
    const int i = blockIdx.x * blockDim.x + threadIdx.x;   // 65536
    const float nw = ema_weight[i] * DECAY + sums[i] * OMD;
    out_weight[i]   = nw;
    out_codebook[i] = nw / new_count[i >> 6];
}

// ---------------------------------------------------------------------------
extern "C" void kernel_launch(void* const* d_in, const int* in_sizes, int n_in,
                              void* d_out, int out_size, void* d_ws, size_t ws_size,
                              hipStream_t stream) {
    const float* x          = (const float*)d_in[0];
    const float* cb         = (const float*)d_in[1];
    const float* ema_count  = (const float*)d_in[2];
    const float* ema_weight = (const float*)d_in[3];

    // d_out flat layout, reference return order:
    float* out          = (float*)d_out;
    float* disc         = out;                                        // 131072*1024
    float* quant        = disc + (size_t)N_ROWS * NUM_EMB;            // 131072*64
    float* out_count    = quant + (size_t)N_ROWS * EMB_DIM;           // 1024
    float* out_weight   = out_count + NUM_EMB;                        // 1024*64
    float* out_codebook = out_weight + NUM_EMB * EMB_DIM;             // 1024*64

    // workspace: counts [1024] | sums [1024*64] | cnorm [1024]
    float* ws     = (float*)d_ws;
    float* counts = ws;
    float* sums   = ws + NUM_EMB;
    float* cnorm  = sums + NUM_EMB * EMB_DIM;

    const int zn = NUM_EMB + NUM_EMB * EMB_DIM;
    vq_zero_ws<<<(zn + 255) / 256, 256, 0, stream>>>(ws, zn);
    vq_cnorm<<<(NUM_EMB * 32 + 255) / 256, 256, 0, stream>>>(cb, cnorm);
    // 131072 rows / 16 rows-per-wave / 8 waves-per-block = 1024 blocks
    vq_assign<<<N_ROWS / 16 / 8, 256, 0, stream>>>(x, cb, cnorm, disc, quant,
                                                   counts, sums);
    vq_count<<<1, NUM_EMB, 0, stream>>>(ema_count, counts, out_count);
    vq_weight<<<(NUM_EMB * EMB_DIM) / 256, 256, 0, stream>>>(
        ema_weight, sums, out_count, out_weight, out_codebook);
}